// S4DKernel_67619965108688
// MI455X (gfx1250) — compile-verified
//
#include <hip/hip_runtime.h>
#include <math.h>

typedef __attribute__((ext_vector_type(2))) float v2f;
typedef __attribute__((ext_vector_type(8))) float v8f;

namespace {
constexpr int kH = 1024;
constexpr int kN = 32;          // N/2 complex modes per head
constexpr int kL = 8192;
constexpr int kTile = 256;      // l-values per 16x16 WMMA macro-tile (16*16)
constexpr int kWavesPerH = 4;   // waves cooperating on one head (strided tiles)
constexpr int kTilesPerWave = (kL / kTile) / kWavesPerH;   // 8
constexpr int kAdvance = kWavesPerH * kTile;               // 1024 l-steps between a wave's tiles
constexpr int kBlockThreads = 256;                         // 8 waves per block
}

// K[h, t + 16*i + j] = 2*Re( sum_n Cc[h,n] * w[h,n]^(t+16*i) * w[h,n]^j )
// == (16x64) @ (64x16) real GEMM per tile, via 16x V_WMMA_F32_16X16X4_F32.
__global__ __launch_bounds__(kBlockThreads)
void s4d_vandermonde_wmma(const float* __restrict__ C,           // (H, 32, 2)
                          const float* __restrict__ log_dt,      // (H,)
                          const float* __restrict__ log_A_real,  // (H, 32)
                          const float* __restrict__ A_imag,      // (H, 32)
                          float* __restrict__ K)                 // (H, 8192)
{
  const int wave = blockIdx.x * (kBlockThreads / 32) + ((int)threadIdx.x >> 5);
  const int lane = (int)threadIdx.x & 31;
  const int h = wave / kWavesPerH;
  const int q = wave % kWavesPerH;          // tile phase within the head
  const int mj = lane & 15;                 // A-row (m=i) for this lane; also B-col (j)
  const int o  = (lane & 16) ? 2 : 0;       // upper half-wave owns K-slots +2/+3
  const int mbase = (lane & 16) ? 8 : 0;    // D-matrix: upper lanes hold rows 8..15

  const float dt = expf(log_dt[h]);

  // Fragments (per ISA 16x4/4x16 f32 layouts):
  //   aR[s] : A[i, 4s+o], A[i, 4s+o+1]          = Re(P_n0), Re(P_n1)
  //   aI[s] : A[i, 32+4s+o], A[i, 32+4s+o+1]    = -Im(P_n0), -Im(P_n1)
  //   bR[s] : B[4s+o, j], B[4s+o+1, j]          = Re(Q_n0), Re(Q_n1)
  //   bI[s] : rows 32+4s+o(+1)                  = Im(Q_n0), Im(Q_n1)
  v2f aR[8], aI[8], bR[8], bI[8], zr[8], zi[8];

  const float argA = (float)(q * kTile + 16 * mj);  // initial exponent for this wave's first tile
  const float argB = (float)mj;                     // j for the B fragment
  const float argZ = (float)kAdvance;               // tile-to-tile advance exponent

#pragma unroll
  for (int s = 0; s < 8; ++s) {
#pragma unroll
    for (int u = 0; u < 2; ++u) {
      const int n = 4 * s + o + u;
      const float lar = log_A_real[h * kN + n];
      const float ai  = A_imag[h * kN + n];
      const float Cr  = C[(h * kN + n) * 2 + 0];
      const float Ci  = C[(h * kN + n) * 2 + 1];

      const float ar = -expf(lar);          // Re(A)
      const float da = dt * ar;             // Re(dtA)
      const float db = dt * ai;             // Im(dtA)

      // w = exp(dtA)
      float sw, cw; sincosf(db, &sw, &cw);
      const float ew = expf(da);
      const float wr = ew * cw, wi = ew * sw;

      // (w - 1) / A  via conj(A)/|A|^2
      const float inv_m2 = 1.0f / (ar * ar + ai * ai);
      const float gr = ((wr - 1.0f) * ar + wi * ai) * inv_m2;
      const float gi = (wi * ar - (wr - 1.0f) * ai) * inv_m2;

      // 2 * Cc  (fold the 2x real-part factor here)
      const float ccr = 2.0f * (Cr * gr - Ci * gi);
      const float cci = 2.0f * (Cr * gi + Ci * gr);

      // P = 2*Cc * w^(q*256 + 16*i)
      float sA, cA; sincosf(db * argA, &sA, &cA);
      const float eA = expf(da * argA);
      const float Pr = ccr * (eA * cA) - cci * (eA * sA);
      const float Pi = ccr * (eA * sA) + cci * (eA * cA);

      // Q = w^j
      float sB, cB; sincosf(db * argB, &sB, &cB);
      const float eB = expf(da * argB);

      // z = w^1024 (advance between this wave's tiles)
      float sZ, cZ; sincosf(db * argZ, &sZ, &cZ);
      const float eZ = expf(da * argZ);

      aR[s][u] =  Pr;
      aI[s][u] = -Pi;
      bR[s][u] = eB * cB;
      bI[s][u] = eB * sB;
      zr[s][u] = eZ * cZ;
      zi[s][u] = eZ * sZ;
    }
  }

  float* outp = K + h * kL + q * kTile;

#pragma unroll 1
  for (int it = 0; it < kTilesPerWave; ++it) {
    // 16x64x16 real GEMM = 16 chained K=4 f32 WMMAs; two accumulators break
    // the serial D->C dependency so hazard NOPs overlap.
    v8f acc0 = {};
    v8f acc1 = {};
#pragma unroll
    for (int s = 0; s < 8; ++s) {
      acc0 = __builtin_amdgcn_wmma_f32_16x16x4_f32(false, aR[s], false, bR[s],
                                                   (short)0, acc0, false, false);
      acc1 = __builtin_amdgcn_wmma_f32_16x16x4_f32(false, aI[s], false, bI[s],
                                                   (short)0, acc1, false, false);
    }

    // D[m, j] -> K[h, base + 16*m + j]; lanes 0..15 of each VGPR are contiguous.
#pragma unroll
    for (int v = 0; v < 8; ++v) {
      outp[16 * (v + mbase) + mj] = acc0[v] + acc1[v];
    }

    // Advance A-fragment: P *= z (complex). Lane holds Re(P) in aR, -Im(P) in aI.
#pragma unroll
    for (int s = 0; s < 8; ++s) {
      const v2f pr  = aR[s];
      const v2f npi = aI[s];
      aR[s] = pr * zr[s] + npi * zi[s];   // Re(P*z)   (Pi = -npi)
      aI[s] = npi * zr[s] - pr * zi[s];   // -Im(P*z)
    }
    outp += kAdvance;
  }
}

extern "C" void kernel_launch(void* const* d_in, const int* in_sizes, int n_in,
                              void* d_out, int out_size, void* d_ws, size_t ws_size,
                              hipStream_t stream) {
  const float* C          = (const float*)d_in[0];  // (1024, 32, 2)
  const float* log_dt     = (const float*)d_in[1];  // (1024,)
  const float* log_A_real = (const float*)d_in[2];  // (1024, 32)
  const float* A_imag     = (const float*)d_in[3];  // (1024, 32)
  // d_in[4] = L (scalar), compile-time constant 8192 here.
  float* out = (float*)d_out;                       // (1024, 8192)

  const int totalWaves = kH * kWavesPerH;                      // 4096
  const int blocks = totalWaves / (kBlockThreads / 32);        // 512
  s4d_vandermonde_wmma<<<blocks, kBlockThreads, 0, stream>>>(
      C, log_dt, log_A_real, A_imag, out);
}